// CosmicGNN_9878424781268
// MI455X (gfx1250) — compile-verified
//
#include <hip/hip_runtime.h>
#include <math.h>

typedef float v2f __attribute__((ext_vector_type(2)));
typedef float v8f __attribute__((ext_vector_type(8)));

#define NGRAPHS 64
#define H 32

// ---------------- degree / normalization ----------------

__global__ void k_init_deg(float* __restrict__ deg, int n) {
    int i = blockIdx.x * blockDim.x + threadIdx.x;
    if (i < n) deg[i] = 1.0f;  // self loop
}

__global__ void k_deg_edges(const long long* __restrict__ dst,
                            float* __restrict__ deg, int E) {
    int e = blockIdx.x * blockDim.x + threadIdx.x;
    if (e < E) atomicAdd(&deg[(int)dst[e]], 1.0f);
}

__global__ void k_inv_sqrt(float* __restrict__ deg, int n) {
    int i = blockIdx.x * blockDim.x + threadIdx.x;
    if (i < n) deg[i] = rsqrtf(fmaxf(deg[i], 1.0f));
}

// ---------------- dense transform: Y[N x 32] = X[N x K] @ W[K x 32] ----------------
// fp32 WMMA 16x16x4: one wave handles a 16-row tile, two 16x16 C tiles.
// A layout (16x4 f32): lanes 0-15 M=0..15 {K=kb,kb+1}; lanes 16-31 {K=kb+2,kb+3}.
// B layout (4x16 f32): lanes 0-15 N=0..15 {K=kb,kb+1}; lanes 16-31 {K=kb+2,kb+3}.
// C layout (16x16 f32): VGPR v, lanes 0-15 -> M=v, lanes 16-31 -> M=v+8.
__global__ void k_gemm_wmma(const float* __restrict__ X, const float* __restrict__ W,
                            float* __restrict__ Y, int nrows, int K) {
    int wave = (int)((blockIdx.x * blockDim.x + threadIdx.x) >> 5);
    int lane = threadIdx.x & 31;
    int row0 = wave * 16;
    if (row0 >= nrows) return;          // wave-uniform: EXEC stays all-ones for WMMA
    int m  = lane & 15;
    int hi = lane >> 4;                 // 0 or 1 (lane half)
    const float* xr = X + (size_t)(row0 + m) * K;
    v8f c0 = {}; v8f c1 = {};
    for (int kb = 0; kb < K; kb += 4) {
        int ka = kb + hi * 2;
        v2f a;  a.x  = xr[ka];               a.y  = xr[ka + 1];
        v2f b0; b0.x = W[ka * H + m];        b0.y = W[(ka + 1) * H + m];
        v2f b1; b1.x = W[ka * H + 16 + m];   b1.y = W[(ka + 1) * H + 16 + m];
        c0 = __builtin_amdgcn_wmma_f32_16x16x4_f32(false, a, false, b0, (short)0, c0, false, false);
        c1 = __builtin_amdgcn_wmma_f32_16x16x4_f32(false, a, false, b1, (short)0, c1, false, false);
    }
    float* yr = Y + (size_t)row0 * H;
#pragma unroll
    for (int v = 0; v < 8; ++v) {
        int rr = v + hi * 8;
        yr[(size_t)rr * H + m]      = c0[v];
        yr[(size_t)rr * H + 16 + m] = c1[v];
    }
}

// ---------------- edge scatter-aggregate: agg[d] += xw[s] * inv[s]*inv[d] ----------------
// 8 threads per edge, 4 floats each (float4 gather, 4 atomics).
__global__ void k_edge_agg(const long long* __restrict__ src, const long long* __restrict__ dst,
                           const float* __restrict__ inv, const float* __restrict__ xw,
                           float* __restrict__ agg, int E) {
    int gid = blockIdx.x * blockDim.x + threadIdx.x;
    int e = gid >> 3;
    if (e >= E) return;
    int q = (gid & 7) * 4;
    int s = (int)src[e], d = (int)dst[e];
    float coef = inv[s] * inv[d];
    const float4 v = *(const float4*)(xw + (size_t)s * H + q);
    float* o = agg + (size_t)d * H + q;
    atomicAdd(o + 0, v.x * coef);
    atomicAdd(o + 1, v.y * coef);
    atomicAdd(o + 2, v.z * coef);
    atomicAdd(o + 3, v.w * coef);
}

// ---------------- layer-1 finalize: self-loop + bias + relu + Rosehip gate ----------------
// One wave per node (32 lanes = 32 features); gate dot via wave shuffle reduction.
__global__ void k_finalize_gate(const float* __restrict__ xw, float* __restrict__ agg,
                                const float* __restrict__ inv, const float* __restrict__ b,
                                const float* __restrict__ Wg, const float* __restrict__ bg, int n) {
    int node = (int)((blockIdx.x * blockDim.x + threadIdx.x) >> 5);
    int f = threadIdx.x & 31;
    if (node >= n) return;
    float iv = inv[node];
    size_t off = (size_t)node * H + f;
    float val = agg[off] + xw[off] * iv * iv + b[f];
    val = fmaxf(val, 0.0f);
    float g = val * Wg[f];
#pragma unroll
    for (int o = 16; o > 0; o >>= 1) g += __shfl_xor(g, o, 32);
    g = 1.0f / (1.0f + __expf(-(g + bg[0])));
    agg[off] = val * g;
}

// ---------------- generic finalize: self-loop + bias (+ optional relu) ----------------
__global__ void k_finalize(const float* __restrict__ xw, float* __restrict__ agg,
                           const float* __restrict__ inv, const float* __restrict__ b,
                           int n, int do_relu) {
    int gid = blockIdx.x * blockDim.x + threadIdx.x;
    if (gid >= n * H) return;
    int node = gid >> 5;
    int f = gid & 31;
    float iv = inv[node];
    float val = agg[gid] + xw[gid] * iv * iv + b[f];
    if (do_relu) val = fmaxf(val, 0.0f);
    agg[gid] = val;
}

// ---------------- per-graph mean pool (LDS-staged, batch is sorted) ----------------
__global__ void k_pool(const float* __restrict__ h, const long long* __restrict__ batch,
                       float* __restrict__ pool, float* __restrict__ cnt,
                       int n, int nodes_per_block) {
    __shared__ float sp[NGRAPHS * H];
    __shared__ float sc[NGRAPHS];
    for (int i = threadIdx.x; i < NGRAPHS * H; i += blockDim.x) sp[i] = 0.0f;
    for (int i = threadIdx.x; i < NGRAPHS; i += blockDim.x) sc[i] = 0.0f;
    __syncthreads();
    int f  = threadIdx.x & 31;
    int ln = threadIdx.x >> 5;   // 0..7
    int base = blockIdx.x * nodes_per_block;
    for (int i = ln; i < nodes_per_block; i += 8) {
        int node = base + i;
        if (node >= n) break;
        int bg = (int)batch[node];
        atomicAdd(&sp[bg * H + f], h[(size_t)node * H + f]);
        if (f == 0) atomicAdd(&sc[bg], 1.0f);
    }
    __syncthreads();
    for (int i = threadIdx.x; i < NGRAPHS * H; i += blockDim.x) atomicAdd(&pool[i], sp[i]);
    for (int i = threadIdx.x; i < NGRAPHS; i += blockDim.x) atomicAdd(&cnt[i], sc[i]);
}

// ---------------- classifier head: mean, 32->64->32->3, softmax ----------------
__global__ void k_head(const float* __restrict__ pool, const float* __restrict__ cnt,
                       const float* __restrict__ Wc1, const float* __restrict__ bc1,
                       const float* __restrict__ Wc2, const float* __restrict__ bc2,
                       const float* __restrict__ Wc3, const float* __restrict__ bc3,
                       float* __restrict__ out) {
    int g = threadIdx.x;
    if (g >= NGRAPHS) return;
    float p[H];
    float c = fmaxf(cnt[g], 1.0f);
    for (int f = 0; f < H; ++f) p[f] = pool[g * H + f] / c;
    float z1[64];
    for (int j = 0; j < 64; ++j) {
        float s = bc1[j];
        for (int f = 0; f < H; ++f) s += p[f] * Wc1[f * 64 + j];
        z1[j] = fmaxf(s, 0.0f);
    }
    float z2[32];
    for (int j = 0; j < 32; ++j) {
        float s = bc2[j];
        for (int k = 0; k < 64; ++k) s += z1[k] * Wc2[k * 32 + j];
        z2[j] = fmaxf(s, 0.0f);
    }
    float z3[3];
    for (int j = 0; j < 3; ++j) {
        float s = bc3[j];
        for (int k = 0; k < 32; ++k) s += z2[k] * Wc3[k * 3 + j];
        z3[j] = s;
    }
    float mx = fmaxf(z3[0], fmaxf(z3[1], z3[2]));
    float e0 = __expf(z3[0] - mx), e1 = __expf(z3[1] - mx), e2 = __expf(z3[2] - mx);
    float den = e0 + e1 + e2;
    out[g * 3 + 0] = e0 / den;
    out[g * 3 + 1] = e1 / den;
    out[g * 3 + 2] = e2 / den;
}

extern "C" void kernel_launch(void* const* d_in, const int* in_sizes, int n_in,
                              void* d_out, int out_size, void* d_ws, size_t ws_size,
                              hipStream_t stream) {
    const float*     x     = (const float*)d_in[0];
    const long long* ei    = (const long long*)d_in[1];
    const long long* batch = (const long long*)d_in[2];
    const float* W1  = (const float*)d_in[3];
    const float* b1  = (const float*)d_in[4];
    const float* Wg  = (const float*)d_in[5];
    const float* bg  = (const float*)d_in[6];
    const float* W2  = (const float*)d_in[7];
    const float* b2  = (const float*)d_in[8];
    const float* W3  = (const float*)d_in[9];
    const float* b3  = (const float*)d_in[10];
    const float* Wc1 = (const float*)d_in[11];
    const float* bc1 = (const float*)d_in[12];
    const float* Wc2 = (const float*)d_in[13];
    const float* bc2 = (const float*)d_in[14];
    const float* Wc3 = (const float*)d_in[15];
    const float* bc3 = (const float*)d_in[16];

    int n = in_sizes[0] / 16;
    int E = in_sizes[1] / 2;
    const long long* src = ei;
    const long long* dst = ei + E;

    char* ws = (char*)d_ws;
    size_t fb = (size_t)n * H * sizeof(float);
    float* A    = (float*)ws; ws += fb;                       // xw scratch
    float* B    = (float*)ws; ws += fb;                       // h1 / agg3
    float* C    = (float*)ws; ws += fb;                       // h2 / agg2
    float* inv  = (float*)ws; ws += (size_t)n * sizeof(float);
    float* pool = (float*)ws; ws += NGRAPHS * H * sizeof(float);
    float* cnt  = (float*)ws; ws += NGRAPHS * sizeof(float);

    const int t = 256;
    hipMemsetAsync(B, 0, fb, stream);                              // agg buffer, layer 1
    hipMemsetAsync(C, 0, fb, stream);                              // agg buffer, layer 2
    hipMemsetAsync(pool, 0, (NGRAPHS * H + NGRAPHS) * sizeof(float), stream);

    // symmetric normalization (computed once, reused for all 3 convs)
    k_init_deg<<<(n + t - 1) / t, t, 0, stream>>>(inv, n);
    k_deg_edges<<<(E + t - 1) / t, t, 0, stream>>>(dst, inv, E);
    k_inv_sqrt<<<(n + t - 1) / t, t, 0, stream>>>(inv, n);

    int tiles = (n + 15) / 16;
    int gemm_blocks = (tiles + 7) / 8;                 // 8 waves per 256-thread block
    int edge_blocks = (int)(((long long)E * 8 + t - 1) / t);
    int nf_blocks   = (n * H + t - 1) / t;

    // layer 1: h1 = gate(relu(GCN(x, W1) + b1))
    k_gemm_wmma<<<gemm_blocks, t, 0, stream>>>(x, W1, A, n, 16);
    k_edge_agg<<<edge_blocks, t, 0, stream>>>(src, dst, inv, A, B, E);
    k_finalize_gate<<<nf_blocks, t, 0, stream>>>(A, B, inv, b1, Wg, bg, n);

    // layer 2: h2 = relu(GCN(h1, W2) + b2)
    k_gemm_wmma<<<gemm_blocks, t, 0, stream>>>(B, W2, A, n, 32);
    hipMemsetAsync(B, 0, fb, stream);                              // recycle B as layer-3 agg
    k_edge_agg<<<edge_blocks, t, 0, stream>>>(src, dst, inv, A, C, E);
    k_finalize<<<nf_blocks, t, 0, stream>>>(A, C, inv, b2, n, 1);

    // layer 3: h3 = GCN(h2, W3) + b3
    k_gemm_wmma<<<gemm_blocks, t, 0, stream>>>(C, W3, A, n, 32);
    k_edge_agg<<<edge_blocks, t, 0, stream>>>(src, dst, inv, A, B, E);
    k_finalize<<<nf_blocks, t, 0, stream>>>(A, B, inv, b3, n, 0);

    // per-graph mean pool + MLP head + softmax
    const int npb = 1024;
    k_pool<<<(n + npb - 1) / npb, t, 0, stream>>>(B, batch, pool, cnt, n, npb);
    k_head<<<1, 64, 0, stream>>>(pool, cnt, Wc1, bc1, Wc2, bc2, Wc3, bc3, (float*)d_out);
}